// DeformableElementwiseDynamicConv2d_76149770158460
// MI455X (gfx1250) — compile-verified
//
#include <hip/hip_runtime.h>

typedef __attribute__((ext_vector_type(2))) float v2f;
typedef __attribute__((ext_vector_type(8))) float v8f;

#define H_   128
#define W_   128
#define HW_  16384         // H_*W_
#define CI   64
#define KK   9
#define KTOT 576           // CI * KK
#define NSP  144           // 16 pixels * 9 taps
#define LDA  580           // padded LDS row stride (floats): 580%64=4 -> conflict-free b64

// ---------------------------------------------------------------------------
// Weight repack:  w[n][c][ky][kx]  ->  wt[n][t*64+c]   (tap-major K order)
// ---------------------------------------------------------------------------
__global__ __launch_bounds__(256)
void repack_w(const float* __restrict__ w, float* __restrict__ wt, int total)
{
    int i = blockIdx.x * 256 + threadIdx.x;
    if (i < total) {
        int n = i / KTOT, r = i - n * KTOT;
        int t = r >> 6, c = r & 63;
        wt[i] = w[n * KTOT + c * KK + t];
    }
}

// ---------------------------------------------------------------------------
// 3x3 / pad1 / stride1 conv as implicit GEMM, tap-major K order.
// One wave per (16 row-contiguous pixels) x (16 out channels) tile.
// grid: x = H*(W/16) = 1024 tiles, y = ceil(Co/16), z = B. block = 32.
// ---------------------------------------------------------------------------
__global__ __launch_bounds__(32)
void conv3x3_wmma(const float* __restrict__ x, const float* __restrict__ wt,
                  float* __restrict__ out, int Co, int relu)
{
    const int lane = threadIdx.x;
    const int tile = blockIdx.x;
    const int h    = tile >> 3;            // 8 tiles per image row
    const int w0   = (tile & 7) << 4;
    const int b    = blockIdx.z;

    const int mrow = lane & 15;            // A row (pixel) / B col (out channel)
    const int khi  = (lane >> 4) << 1;     // K sub-offset per ISA A layout: 0 or 2
    const int n    = (blockIdx.y << 4) + mrow;
    const int ncl  = min(n, Co - 1);       // clamped: garbage col never stored

    const float* __restrict__ wtrow = wt + (size_t)ncl * KTOT;
    const float* __restrict__ xb    = x + (size_t)b * CI * HW_;

    v8f acc = {};
#pragma unroll
    for (int t = 0; t < KK; ++t) {
        const int ky = t / 3, kx = t - ky * 3;           // compile-time
        const int y  = h + ky - 1;
        const int xx = w0 + mrow + kx - 1;
        const bool ok = ((unsigned)y < (unsigned)H_) && ((unsigned)xx < (unsigned)W_);
        const int yc = min(max(y, 0), H_ - 1);
        const int xc = min(max(xx, 0), W_ - 1);
        const float* __restrict__ ap = xb + (yc * W_ + xc);     // safe base, c=0
        const float* __restrict__ bp = wtrow + t * CI + khi;    // contiguous in c
#pragma unroll
        for (int ci = 0; ci < 16; ++ci) {
            const int c0 = (ci << 2) + khi;
            float a0 = ap[(size_t)c0 * HW_];           // literal-offset loads
            float a1 = ap[(size_t)(c0 + 1) * HW_];
            v2f a;
            a.x = ok ? a0 : 0.0f;
            a.y = ok ? a1 : 0.0f;
            float2 wv = *(const float2*)(bp + (ci << 2));
            v2f bb; bb.x = wv.x; bb.y = wv.y;
            acc = __builtin_amdgcn_wmma_f32_16x16x4_f32(false, a, false, bb,
                                                        (short)0, acc, false, false);
        }
    }

    if (n < Co) {
        const int pbase = (lane >> 4) << 3;                  // 0 or 8
        float* __restrict__ orow =
            out + (((size_t)b * Co + n) * H_ + h) * W_ + w0 + pbase;
        float r[8];
#pragma unroll
        for (int i = 0; i < 8; ++i) {
            float v = acc[i];
            r[i] = relu ? fmaxf(v, 0.0f) : v;
        }
        *(float4*)(orow)     = make_float4(r[0], r[1], r[2], r[3]);
        *(float4*)(orow + 4) = make_float4(r[4], r[5], r[6], r[7]);
    }
}

// ---------------------------------------------------------------------------
// Fused: 2x deformable conv (bilinear gather -> LDS A tile -> WMMA K-loop)
//        + softmax over DYNAMIC axis of att + weighted accumulate.
// block = 128 (4 waves share one 16-pixel A tile; each wave owns 16 out chans)
// grid: x = 1024 tiles, y = B.   wdt is the repacked (2,64,[t*64+c]) weights.
// ---------------------------------------------------------------------------
__global__ __launch_bounds__(128)
void deform_fuse_wmma(const float* __restrict__ x,
                      const float* __restrict__ offsets,  // (B,36,H,W)
                      const float* __restrict__ att,      // (B,128,H,W)
                      const float* __restrict__ wdt,      // repacked (2*64, 576)
                      float* __restrict__ out)            // (B,64,H,W)
{
    __shared__ float sA[16 * LDA];     // A tile, k = t*64+c, padded rows
    __shared__ int4   sO[NSP];         // 4 clamped corner offsets per sample pt
    __shared__ float4 sW[NSP];         // 4 bilinear weights (0 when invalid)

    const int tid  = threadIdx.x;
    const int lane = tid & 31;
    const int wv   = tid >> 5;
    const int tile = blockIdx.x;
    const int b    = blockIdx.y;
    const int h    = tile >> 3;
    const int w0   = (tile & 7) << 4;

    const int mrow = lane & 15;
    const int khi  = (lane >> 4) << 1;
    const int n    = (wv << 4) + mrow;     // out channel 0..63

    const float* __restrict__ xb = x + (size_t)b * CI * HW_;

    v8f facc[2];
#pragma unroll
    for (int dk = 0; dk < 2; ++dk) {
        // ---- phase 1: bilinear metadata (validity folded into weights)
        if (tid < NSP) {
            int m = tid & 15, t = tid >> 4;
            int ky = t / 3, kx = t - ky * 3;
            const float* __restrict__ ob =
                offsets + (((size_t)b * 36 + dk * 18 + t * 2) * H_ + h) * W_ + w0 + m;
            float offy = ob[0];
            float offx = ob[HW_];
            float py = (float)(h - 1 + ky) + offy;
            float px = (float)(w0 + m - 1 + kx) + offx;
            float fy = floorf(py), fx = floorf(px);
            int y0 = (int)fy, x0 = (int)fx;
            float ay = py - fy, ax = px - fx;
            bool vy0 = (unsigned)y0 < (unsigned)H_;
            bool vy1 = (unsigned)(y0 + 1) < (unsigned)H_;
            bool vx0 = (unsigned)x0 < (unsigned)W_;
            bool vx1 = (unsigned)(x0 + 1) < (unsigned)W_;
            int yc0 = min(max(y0, 0), H_ - 1),     xc0 = min(max(x0, 0), W_ - 1);
            int yc1 = min(max(y0 + 1, 0), H_ - 1), xc1 = min(max(x0 + 1, 0), W_ - 1);
            int4 o;
            o.x = yc0 * W_ + xc0;  o.y = yc0 * W_ + xc1;
            o.z = yc1 * W_ + xc0;  o.w = yc1 * W_ + xc1;
            float4 wt4;
            wt4.x = (vy0 && vx0) ? (1.0f - ay) * (1.0f - ax) : 0.0f;
            wt4.y = (vy0 && vx1) ? (1.0f - ay) * ax          : 0.0f;
            wt4.z = (vy1 && vx0) ? ay * (1.0f - ax)          : 0.0f;
            wt4.w = (vy1 && vx1) ? ay * ax                   : 0.0f;
            sO[tid] = o;  sW[tid] = wt4;
        }
        __syncthreads();
        // ---- phase 2: gather 64ch x 144 sample pts into LDS (no divisions)
        {
            int sp = tid, c = 0;                 // idx = tid + 128*iter
#pragma unroll 4
            for (int iter = 0; iter < (CI * NSP) / 128; ++iter) {
                int m = sp & 15, t = sp >> 4;
                int4 o = sO[sp];  float4 g = sW[sp];
                const float* __restrict__ xc = xb + (size_t)c * HW_;
                float v = xc[o.x] * g.x + xc[o.y] * g.y
                        + xc[o.z] * g.z + xc[o.w] * g.w;
                sA[m * LDA + t * CI + c] = v;
                sp += 128; if (sp >= NSP) { sp -= NSP; ++c; }
            }
        }
        __syncthreads();
        // ---- phase 3: WMMA K-loop, A from LDS (ds_load_b64), B contiguous
        const float* __restrict__ wrow = wdt + ((size_t)(dk * 64 + n)) * KTOT;
        const float* __restrict__ aBase = &sA[mrow * LDA];
        v8f acc = {};
#pragma unroll
        for (int it = 0; it < KTOT / 4; ++it) {
            int k0 = (it << 2) + khi;
            v2f a = *(const v2f*)(aBase + k0);
            float2 wv2 = *(const float2*)(wrow + k0);
            v2f bb; bb.x = wv2.x; bb.y = wv2.y;
            acc = __builtin_amdgcn_wmma_f32_16x16x4_f32(false, a, false, bb,
                                                        (short)0, acc, false, false);
        }
        facc[dk] = acc;
        __syncthreads();   // LDS A tile reused by next dk
    }

    // ---- combine: softmax over dk of att, weighted sum, store
    const int pbase = (lane >> 4) << 3;
    const float* __restrict__ a0p =
        att + (((size_t)b * 128 + n) * H_ + h) * W_ + w0 + pbase;
    const float* __restrict__ a1p = a0p + (size_t)64 * HW_;
    float* __restrict__ orow =
        out + (((size_t)b * 64 + n) * H_ + h) * W_ + w0 + pbase;

    float res[8];
#pragma unroll
    for (int r = 0; r < 8; ++r) {
        float u0 = a0p[r], u1 = a1p[r];
        float mx = fmaxf(u0, u1);
        float e0 = expf(u0 - mx), e1 = expf(u1 - mx);
        float inv = 1.0f / (e0 + e1);
        res[r] = facc[0][r] * (e0 * inv) + facc[1][r] * (e1 * inv);
    }
    *(float4*)(orow)     = make_float4(res[0], res[1], res[2], res[3]);
    *(float4*)(orow + 4) = make_float4(res[4], res[5], res[6], res[7]);
}

// ---------------------------------------------------------------------------
extern "C" void kernel_launch(void* const* d_in, const int* in_sizes, int n_in,
                              void* d_out, int out_size, void* d_ws, size_t ws_size,
                              hipStream_t stream)
{
    (void)in_sizes; (void)n_in; (void)out_size; (void)ws_size;
    const float* x  = (const float*)d_in[0];
    const float* w1 = (const float*)d_in[1];
    const float* w2 = (const float*)d_in[2];
    const float* w3 = (const float*)d_in[3];
    const float* wa = (const float*)d_in[4];
    const float* wd = (const float*)d_in[5];
    float* out = (float*)d_out;

    const int B = 8;
    const size_t img = (size_t)CI * HW_;             // 1,048,576 floats
    float* m1   = (float*)d_ws;                      // B*64*H*W
    float* m2   = m1 + (size_t)B * img;              // B*64*H*W
    float* offs = m2 + (size_t)B * img;              // B*36*H*W
    float* attb = offs + (size_t)B * 36 * HW_;       // B*128*H*W
    float* w1t  = attb + (size_t)B * 128 * HW_;      // 64*576
    float* w2t  = w1t + (size_t)64 * KTOT;
    float* w3t  = w2t + (size_t)64 * KTOT;           // 36*576
    float* wat  = w3t + (size_t)36 * KTOT;           // 128*576
    float* wdt  = wat + (size_t)128 * KTOT;          // 128*576 (2 x 64)

    // repack all weights to tap-major K order
    repack_w<<<dim3((64  * KTOT + 255) / 256), 256, 0, stream>>>(w1, w1t, 64  * KTOT);
    repack_w<<<dim3((64  * KTOT + 255) / 256), 256, 0, stream>>>(w2, w2t, 64  * KTOT);
    repack_w<<<dim3((36  * KTOT + 255) / 256), 256, 0, stream>>>(w3, w3t, 36  * KTOT);
    repack_w<<<dim3((128 * KTOT + 255) / 256), 256, 0, stream>>>(wa, wat, 128 * KTOT);
    repack_w<<<dim3((128 * KTOT + 255) / 256), 256, 0, stream>>>(wd, wdt, 128 * KTOT);

    dim3 blk(32);
    conv3x3_wmma<<<dim3(1024, 4, B), blk, 0, stream>>>(x,  w1t, m1,  64, 1);
    conv3x3_wmma<<<dim3(1024, 4, B), blk, 0, stream>>>(m1, w2t, m2,  64, 1);
    conv3x3_wmma<<<dim3(1024, 3, B), blk, 0, stream>>>(m2, w3t, offs, 36, 0);
    conv3x3_wmma<<<dim3(1024, 8, B), blk, 0, stream>>>(x,  wat, attb, 128, 0);
    deform_fuse_wmma<<<dim3(1024, B), dim3(128), 0, stream>>>(x, offs, attb, wdt, out);
}